// SetAbstraction_86517821210640
// MI455X (gfx1250) — compile-verified
//
#include <hip/hip_runtime.h>
#include <math.h>

// ---- problem constants (match reference) ----
#define B_CLOUDS 8
#define N_PER    4096
#define M_PER    2048
#define NSAMPLE  32
#define D_FEAT   64
#define C_IN     67
#define C_PAD    68              // pad K to multiple of 4 for 16x16x4 WMMA
#define RADIUS2  0.01f           // 0.1^2
#define BN_EPSF  1e-5f
#define N_CENTERS (B_CLOUDS * M_PER)   // 16384

typedef __attribute__((ext_vector_type(2))) float v2f;
typedef __attribute__((ext_vector_type(4))) float v4f;
typedef __attribute__((ext_vector_type(8))) float v8f;

__device__ __forceinline__ void lexmax(float &v, int &i, float ov, int oi) {
  if (ov > v || (ov == v && oi < i)) { v = ov; i = oi; }
}
__device__ __forceinline__ void lexmin(float &v, int &i, float ov, int oi) {
  if (ov < v || (ov == v && oi < i)) { v = ov; i = oi; }
}

// ============================================================
// Kernel 1: farthest point sampling. One 1024-thread WG (32 wave32) per cloud.
// Each thread owns 4 points (pos cached in registers, mind in registers).
// Also writes pos_new / batch_new portions of d_out.
// ============================================================
__global__ void __launch_bounds__(1024)
fps_kernel(const float* __restrict__ pos, int* __restrict__ fps_idx,
           float* __restrict__ out) {
  __shared__ float sv[32];
  __shared__ int   si[32];
  __shared__ float sP[3];
  const int b    = blockIdx.x;
  const int tid  = threadIdx.x;
  const int lane = tid & 31, w = tid >> 5;
  const int cb   = b * N_PER;

  float px[4], py[4], pz[4], mind[4];
#pragma unroll
  for (int q = 0; q < 4; ++q) {
    int j = tid + q * 1024;
    px[q] = pos[(cb + j) * 3 + 0];
    py[q] = pos[(cb + j) * 3 + 1];
    pz[q] = pos[(cb + j) * 3 + 2];
    mind[q] = __builtin_inff();
  }
  float* out_pos   = out + (size_t)N_CENTERS * 128;
  float* out_batch = out_pos + (size_t)N_CENTERS * 3;

  for (int m = 0; m < M_PER; ++m) {
    // local argmax (ties -> lowest index, matching jnp.argmax)
    float lv = mind[0]; int li = tid;
#pragma unroll
    for (int q = 1; q < 4; ++q)
      if (mind[q] > lv) { lv = mind[q]; li = tid + q * 1024; }
    // wave32 reduction
#pragma unroll
    for (int off = 16; off; off >>= 1) {
      float ov = __shfl_xor(lv, off);
      int   oi = __shfl_xor(li, off);
      lexmax(lv, li, ov, oi);
    }
    if (lane == 0) { sv[w] = lv; si[w] = li; }
    __syncthreads();
    if (w == 0) {
      float v = sv[lane]; int i = si[lane];
#pragma unroll
      for (int off = 16; off; off >>= 1) {
        float ov = __shfl_xor(v, off);
        int   oi = __shfl_xor(i, off);
        lexmax(v, i, ov, oi);
      }
      if (lane == 0) {
        float cx = pos[(cb + i) * 3 + 0];
        float cy = pos[(cb + i) * 3 + 1];
        float cz = pos[(cb + i) * 3 + 2];
        sP[0] = cx; sP[1] = cy; sP[2] = cz;
        int g = b * M_PER + m;
        fps_idx[g] = i;
        out_pos[g * 3 + 0] = cx;
        out_pos[g * 3 + 1] = cy;
        out_pos[g * 3 + 2] = cz;
        out_batch[g] = (float)b;   // int cloud id stored numerically
      }
    }
    __syncthreads();
    float cx = sP[0], cy = sP[1], cz = sP[2];
#pragma unroll
    for (int q = 0; q < 4; ++q) {
      float dx = px[q] - cx, dy = py[q] - cy, dz = pz[q] - cz;
      mind[q] = fminf(mind[q], dx * dx + dy * dy + dz * dz);
    }
  }
}

// ============================================================
// Kernel 2: ball query. 256 threads = 8 waves, 1 wave per center.
// Cloud positions staged in LDS; per-wave d2 scratch in LDS; iterative
// lexicographic-min selection (32 rounds). Only the winning lane rescans
// its 128 values each round. Invalid slots duplicate the center point
// (max-aggregation-invariant substitute for the -inf mask).
// Dynamic LDS: (3*4096 + 8*4096) floats.
// ============================================================
__global__ void __launch_bounds__(256)
ballq_kernel(const float* __restrict__ pos, const int* __restrict__ fps_idx,
             int* __restrict__ nbr) {
  extern __shared__ char smem[];
  float* sx = (float*)smem;
  float* sy = sx + N_PER;
  float* sz = sy + N_PER;
  float* sd = sz + N_PER;          // 8 * 4096
  const int tid = threadIdx.x, lane = tid & 31, w = tid >> 5;
  const int b  = blockIdx.x >> 8;  // 8 centers/block, 2048 centers/cloud
  const int cb = b * N_PER;

  for (int j = tid; j < N_PER; j += 256) {
    sx[j] = pos[(cb + j) * 3 + 0];
    sy[j] = pos[(cb + j) * 3 + 1];
    sz[j] = pos[(cb + j) * 3 + 2];
  }
  __syncthreads();

  const int cg  = blockIdx.x * 8 + w;
  const int ctr = fps_idx[cg];
  const float cx = sx[ctr], cy = sy[ctr], cz = sz[ctr];
  volatile float* d2 = sd + w * N_PER;

  float lv = __builtin_inff(); int lj = N_PER;
  for (int q = 0; q < N_PER / 32; ++q) {     // lane-interleaved, bank-friendly
    int j = q * 32 + lane;
    float dx = sx[j] - cx, dy = sy[j] - cy, dz = sz[j] - cz;
    float v = dx * dx + dy * dy + dz * dz;
    d2[j] = v;
    if (v < lv) { lv = v; lj = j; }
  }
  int* outp = nbr + cg * NSAMPLE;
  for (int r = 0; r < NSAMPLE; ++r) {
    float v = lv; int j = lj;
#pragma unroll
    for (int off = 16; off; off >>= 1) {
      float ov = __shfl_xor(v, off);
      int   oj = __shfl_xor(j, off);
      lexmin(v, j, ov, oj);
    }
    if (v <= RADIUS2) {
      if (lane == 0) outp[r] = cb + j;       // absolute point index
      if ((j & 31) == lane) {                // owner removes + rescans
        d2[j] = __builtin_inff();
        lv = __builtin_inff(); lj = N_PER;
        for (int q = 0; q < N_PER / 32; ++q) {
          int jj = q * 32 + lane;
          float vv = d2[jj];
          if (vv < lv) { lv = vv; lj = jj; }
        }
      }
    } else {
      if (lane == 0) outp[r] = cb + ctr;     // duplicate center: max-safe
    }
  }
}

// ============================================================
// Kernel 3: gather + 3-layer MLP (WMMA f32 16x16x4) + BN/ReLU + max-agg.
// 256 threads = 8 waves; wave w owns center (blockIdx*8 + w): 32 rows.
// LDS: X (256x68), Y (256x64), W0^T (64x68), W1^T (64x64), W2^T (128x64),
// folded BN scale/shift. After one barrier, waves run fully independently.
// ============================================================
__global__ void __launch_bounds__(256)
sa_mlp_kernel(const float* __restrict__ h, const float* __restrict__ pos,
              const int* __restrict__ fps_idx, const int* __restrict__ nbr,
              const float* __restrict__ W0, const float* __restrict__ b0,
              const float* __restrict__ g0, const float* __restrict__ be0,
              const float* __restrict__ rm0, const float* __restrict__ rv0,
              const float* __restrict__ W1, const float* __restrict__ b1,
              const float* __restrict__ g1, const float* __restrict__ be1,
              const float* __restrict__ rm1, const float* __restrict__ rv1,
              const float* __restrict__ W2, const float* __restrict__ b2,
              const float* __restrict__ g2, const float* __restrict__ be2,
              const float* __restrict__ rm2, const float* __restrict__ rv2,
              float* __restrict__ out) {
  extern __shared__ char smem[];
  float* sX   = (float*)smem;            // 256 x 68
  float* sY   = sX  + 256 * C_PAD;       // 256 x 64
  float* sW0t = sY  + 256 * 64;          // 64  x 68  ([out][in], padded)
  float* sW1t = sW0t + 64 * C_PAD;       // 64  x 64
  float* sW2t = sW1t + 64 * 64;          // 128 x 64
  float* sS0  = sW2t + 128 * 64;
  float* sT0  = sS0 + 64;
  float* sS1  = sT0 + 64;
  float* sT1  = sS1 + 64;
  float* sS2  = sT1 + 64;
  float* sT2  = sS2 + 128;

  const int tid = threadIdx.x, lane = tid & 31, w = tid >> 5;
  const int kh = lane >> 4;              // K-half select per WMMA A/B layout
  const int lr = lane & 15;

  // --- stage transposed weights + folded BN affine ---
  for (int t = tid; t < 64 * C_PAD; t += 256) {
    int n = t / C_PAD, k = t % C_PAD;
    sW0t[t] = (k < C_IN) ? W0[k * 64 + n] : 0.f;
  }
  for (int t = tid; t < 64 * 64; t += 256) {
    int n = t >> 6, k = t & 63;
    sW1t[t] = W1[k * 64 + n];
  }
  for (int t = tid; t < 128 * 64; t += 256) {
    int n = t >> 6, k = t & 63;
    sW2t[t] = W2[k * 128 + n];
  }
  if (tid < 64) {
    float s = g0[tid] * rsqrtf(rv0[tid] + BN_EPSF);
    sS0[tid] = s; sT0[tid] = (b0[tid] - rm0[tid]) * s + be0[tid];
  } else if (tid < 128) {
    int c = tid - 64;
    float s = g1[c] * rsqrtf(rv1[c] + BN_EPSF);
    sS1[c] = s; sT1[c] = (b1[c] - rm1[c]) * s + be1[c];
  } else {
    int c = tid - 128;
    float s = g2[c] * rsqrtf(rv2[c] + BN_EPSF);
    sS2[c] = s; sT2[c] = (b2[c] - rm2[c]) * s + be2[c];
  }

  // --- gather: lane = sample slot of this wave's center ---
  const int cg  = blockIdx.x * 8 + w;
  const int cld = cg >> 11, cbase = cld * N_PER;
  const int ctr = fps_idx[cg];
  const float cx = pos[(cbase + ctr) * 3 + 0];
  const float cy = pos[(cbase + ctr) * 3 + 1];
  const float cz = pos[(cbase + ctr) * 3 + 2];
  const int sidx = nbr[cg * NSAMPLE + lane];
  float* Xr = sX + (w * 32 + lane) * C_PAD;
#pragma unroll
  for (int c = 0; c < D_FEAT; c += 4)
    *(v4f*)(Xr + c) = *(const v4f*)(h + (size_t)sidx * D_FEAT + c);
  Xr[64] = pos[sidx * 3 + 0] - cx;
  Xr[65] = pos[sidx * 3 + 1] - cy;
  Xr[66] = pos[sidx * 3 + 2] - cz;
  Xr[67] = 0.f;
  __syncthreads();   // last cross-wave sync; waves independent from here

  // ---- layer 0: [16x68] @ [68x64], X -> sY ----
  for (int rt = 0; rt < 2; ++rt) {
    const int row0 = w * 32 + rt * 16;
    const float* Ar = sX + (row0 + lr) * C_PAD + 2 * kh;
    v8f acc[4] = {};
    for (int k = 0; k < C_PAD; k += 4) {
      v2f a = *(const v2f*)(Ar + k);
#pragma unroll
      for (int n = 0; n < 4; ++n) {
        v2f bm = *(const v2f*)(sW0t + (n * 16 + lr) * C_PAD + k + 2 * kh);
        acc[n] = __builtin_amdgcn_wmma_f32_16x16x4_f32(
            false, a, false, bm, (short)0, acc[n], false, false);
      }
    }
#pragma unroll
    for (int n = 0; n < 4; ++n) {
      int col = n * 16 + lr;
      float sc = sS0[col], sh = sT0[col];
#pragma unroll
      for (int v = 0; v < 8; ++v) {
        float y = fmaxf(acc[n][v] * sc + sh, 0.f);
        sY[(row0 + v + 8 * kh) * 64 + col] = y;
      }
    }
  }

  // ---- layer 1: [16x64] @ [64x64], sY -> sX (reuse, stride C_PAD) ----
  for (int rt = 0; rt < 2; ++rt) {
    const int row0 = w * 32 + rt * 16;
    const float* Ar = sY + (row0 + lr) * 64 + 2 * kh;
    v8f acc[4] = {};
    for (int k = 0; k < 64; k += 4) {
      v2f a = *(const v2f*)(Ar + k);
#pragma unroll
      for (int n = 0; n < 4; ++n) {
        v2f bm = *(const v2f*)(sW1t + (n * 16 + lr) * 64 + k + 2 * kh);
        acc[n] = __builtin_amdgcn_wmma_f32_16x16x4_f32(
            false, a, false, bm, (short)0, acc[n], false, false);
      }
    }
#pragma unroll
    for (int n = 0; n < 4; ++n) {
      int col = n * 16 + lr;
      float sc = sS1[col], sh = sT1[col];
#pragma unroll
      for (int v = 0; v < 8; ++v) {
        float y = fmaxf(acc[n][v] * sc + sh, 0.f);
        sX[(row0 + v + 8 * kh) * C_PAD + col] = y;
      }
    }
  }

  // ---- layer 2: [16x64] @ [64x128] + in-register max over 32 samples ----
  float runmax[8];
#pragma unroll
  for (int n = 0; n < 8; ++n) runmax[n] = 0.f;   // post-ReLU values >= 0
  for (int rt = 0; rt < 2; ++rt) {
    const int row0 = w * 32 + rt * 16;
    const float* Ar = sX + (row0 + lr) * C_PAD + 2 * kh;
    v8f acc[8] = {};
    for (int k = 0; k < 64; k += 4) {
      v2f a = *(const v2f*)(Ar + k);
#pragma unroll
      for (int n = 0; n < 8; ++n) {
        v2f bm = *(const v2f*)(sW2t + (n * 16 + lr) * 64 + k + 2 * kh);
        acc[n] = __builtin_amdgcn_wmma_f32_16x16x4_f32(
            false, a, false, bm, (short)0, acc[n], false, false);
      }
    }
#pragma unroll
    for (int n = 0; n < 8; ++n) {
      int col = n * 16 + lr;
      float sc = sS2[col], sh = sT2[col];
      float mx = 0.f;
#pragma unroll
      for (int v = 0; v < 8; ++v)
        mx = fmaxf(mx, fmaxf(acc[n][v] * sc + sh, 0.f));
      runmax[n] = fmaxf(runmax[n], mx);
    }
  }
#pragma unroll
  for (int n = 0; n < 8; ++n) {
    float o = fmaxf(runmax[n], __shfl_xor(runmax[n], 16)); // fold row halves
    if (lane < 16) out[(size_t)cg * 128 + n * 16 + lane] = o;
  }
}

// ============================================================
extern "C" void kernel_launch(void* const* d_in, const int* in_sizes, int n_in,
                              void* d_out, int out_size, void* d_ws, size_t ws_size,
                              hipStream_t stream) {
  (void)in_sizes; (void)n_in; (void)out_size; (void)ws_size;
  const float* h   = (const float*)d_in[0];
  const float* pos = (const float*)d_in[1];
  // d_in[2] = batch (implicit, unused)
  const float* W0  = (const float*)d_in[3];
  const float* b0  = (const float*)d_in[4];
  const float* g0  = (const float*)d_in[5];
  const float* be0 = (const float*)d_in[6];
  const float* rm0 = (const float*)d_in[7];
  const float* rv0 = (const float*)d_in[8];
  const float* W1  = (const float*)d_in[9];
  const float* b1  = (const float*)d_in[10];
  const float* g1  = (const float*)d_in[11];
  const float* be1 = (const float*)d_in[12];
  const float* rm1 = (const float*)d_in[13];
  const float* rv1 = (const float*)d_in[14];
  const float* W2  = (const float*)d_in[15];
  const float* b2  = (const float*)d_in[16];
  const float* g2  = (const float*)d_in[17];
  const float* be2 = (const float*)d_in[18];
  const float* rm2 = (const float*)d_in[19];
  const float* rv2 = (const float*)d_in[20];
  float* out = (float*)d_out;

  int* fpsIdx = (int*)d_ws;                 // 16384 ints
  int* nbr    = fpsIdx + N_CENTERS;         // 16384*32 ints

  fps_kernel<<<B_CLOUDS, 1024, 0, stream>>>(pos, fpsIdx, out);

  size_t smemB = (size_t)(N_PER * 3 + 8 * N_PER) * sizeof(float); // 180224 B
  ballq_kernel<<<N_CENTERS / 8, 256, smemB, stream>>>(pos, fpsIdx, nbr);

  size_t smemC = (size_t)(256 * C_PAD + 256 * 64 + 64 * C_PAD + 64 * 64 +
                          128 * 64 + 4 * 64 + 2 * 128) * sizeof(float); // 203776 B
  sa_mlp_kernel<<<N_CENTERS / 8, 256, smemC, stream>>>(
      h, pos, fpsIdx, nbr,
      W0, b0, g0, be0, rm0, rv0,
      W1, b1, g1, be1, rm1, rv1,
      W2, b2, g2, be2, rm2, rv2,
      out);
}